// BatchedMoEOptimized_18451179504159
// MI455X (gfx1250) — compile-verified
//
#include <hip/hip_runtime.h>
#include <hip/hip_bf16.h>

// ---------------------------------------------------------------------------
// MoE FFN (fp32) for gfx1250 using V_WMMA_F32_16X16X4_F32.
//   gate = silu(x @ w1[e]); value = x @ w2[e]; y = (gate*value) @ w3[e]
// Tokens are routed to per-expert 16-row tiles; one block = one tile.
// ---------------------------------------------------------------------------

#define N_EXPERTS 8
#define D_MODEL   1024
#define D_FF      4096
#define N_TOKENS  2048
#define TOP_K     2
#define N_ASSIGN  (N_TOKENS * TOP_K)          // 4096
#define MAX_TILES 264                          // sum ceil(c_e/16) <= 256+8

#define XS_STRIDE 1028                         // LDS row stride (floats), bank-conflict free
#define H_STRIDE  132
#define LDS_BYTES ((16 * XS_STRIDE + 16 * H_STRIDE) * 4)   // 74240 B (<320KB WGP LDS)

typedef float v2f __attribute__((ext_vector_type(2)));
typedef float v8f __attribute__((ext_vector_type(8)));

// --------------------------- routing ---------------------------------------
__global__ void moe_route_kernel(const int* __restrict__ eidx,     // [A]
                                 int* __restrict__ s_tok,          // [A]
                                 int* __restrict__ s_flat,         // [A]
                                 int* __restrict__ tile_e,         // [MAX_TILES]
                                 int* __restrict__ tile_srow,      // [MAX_TILES]
                                 int* __restrict__ tile_nrows,     // [MAX_TILES]
                                 int* __restrict__ n_tiles)        // [1]
{
    __shared__ int cnt[N_EXPERTS];
    __shared__ int off[N_EXPERTS];
    __shared__ int cur[N_EXPERTS];
    const int tid = threadIdx.x;

    if (tid < N_EXPERTS) cnt[tid] = 0;
    __syncthreads();

    for (int a = tid; a < N_ASSIGN; a += 256)
        atomicAdd(&cnt[eidx[a]], 1);
    __syncthreads();

    if (tid == 0) {
        int run = 0, nt = 0;
        for (int e = 0; e < N_EXPERTS; ++e) {
            off[e] = run;
            cur[e] = 0;
            const int c = cnt[e];
            for (int i = 0; i < c; i += 16) {
                tile_e[nt]     = e;
                tile_srow[nt]  = run + i;
                tile_nrows[nt] = (c - i < 16) ? (c - i) : 16;
                ++nt;
            }
            run += c;
        }
        *n_tiles = nt;
    }
    __syncthreads();

    for (int a = tid; a < N_ASSIGN; a += 256) {
        const int e = eidx[a];
        const int p = atomicAdd(&cur[e], 1);
        s_tok[off[e] + p]  = a / TOP_K;
        s_flat[off[e] + p] = a;
    }
}

// --------------------------- fused expert GEMM ------------------------------
// Block: 256 threads = 8 waves. One 16-row tile of one expert.
// Wave w owns output columns [w*128, (w+1)*128): 8 persistent 16x16 f32 tiles.
// All B-matrix accesses use 32-bit offsets from the uniform (SGPR) weight base
// so the backend emits SADDR-form global loads and no 64-bit VALU addressing.
__global__ __launch_bounds__(256, 2)
void moe_expert_kernel(const float* __restrict__ x,
                       const float* __restrict__ w1,
                       const float* __restrict__ w2,
                       const float* __restrict__ w3,
                       const int* __restrict__ s_tok,
                       const int* __restrict__ s_flat,
                       const int* __restrict__ tile_e,
                       const int* __restrict__ tile_srow,
                       const int* __restrict__ tile_nrows,
                       const int* __restrict__ n_tiles,
                       float* __restrict__ y)             // [A, D_MODEL]
{
    const int bid = blockIdx.x;
    if (bid >= *n_tiles) return;                 // uniform: whole wave exits

    const int e     = tile_e[bid];
    const int srow  = tile_srow[bid];
    const int nrows = tile_nrows[bid];

    extern __shared__ float lds[];
    float* __restrict__ xs  = lds;                    // [16][XS_STRIDE]
    float* __restrict__ hid = lds + 16 * XS_STRIDE;   // [16][H_STRIDE]
    __shared__ int stk[16];                           // token id per tile row
    __shared__ int sfl[16];                           // flat assignment id (-1 = pad)

    const int tid  = threadIdx.x;
    const int wave = tid >> 5;        // 0..7 (wave32)
    const int lane = tid & 31;
    const int half = lane >> 4;       // 0 or 1
    const int lm   = lane & 15;       // lane mod 16
    const int kr   = half * 2;        // K row base for A/B fragments (ISA 7.12.2)

    const float* __restrict__ W1 = w1 + (size_t)e * D_MODEL * D_FF;
    const float* __restrict__ W2 = w2 + (size_t)e * D_MODEL * D_FF;
    const float* __restrict__ W3 = w3 + (size_t)e * D_FF * D_MODEL;

    if (tid < 16) {
        const int rr = (tid < nrows) ? tid : 0;
        stk[tid] = s_tok[srow + rr];
        sfl[tid] = (tid < nrows) ? s_flat[srow + tid] : -1;
    }
    __syncthreads();

    // Stage the 16 x-rows into LDS (64KB, reused for every d_ff chunk).
    for (int i = tid; i < 16 * D_MODEL; i += 256) {
        const int r = i >> 10;
        const int c = i & (D_MODEL - 1);
        xs[r * XS_STRIDE + c] = x[(unsigned)(stk[r] * D_MODEL + c)];
    }
    __syncthreads();

    v8f acc[8];
#pragma unroll
    for (int t = 0; t < 8; ++t) acc[t] = {};

    const unsigned bcol0 = (unsigned)(wave * 16 + lm);   // phase-1 B column base

    for (int fc = 0; fc < D_FF; fc += 128) {
        // ---- phase 1: gate/value 16x16 tiles, cols f = fc + wave*16 + lm ----
        const unsigned bc1 = (unsigned)fc + bcol0;

        // Prefetch upcoming w3 chunk rows while gemm1 runs (global_prefetch_b8).
        __builtin_prefetch(W3 + (unsigned)((fc + lane) * D_MODEL + wave * 128), 0, 1);
        __builtin_prefetch(W3 + (unsigned)((fc + 32 + lane) * D_MODEL + wave * 128), 0, 1);

        v8f g = {};
        v8f v = {};
        for (int k = 0; k < D_MODEL; k += 4) {
            v2f a;
            a.x = xs[lm * XS_STRIDE + k + kr];
            a.y = xs[lm * XS_STRIDE + k + kr + 1];
            const unsigned o = (unsigned)((k + kr) * D_FF) + bc1;  // <= ~4.2M, 32-bit
            v2f b1, b2;
            b1.x = W1[o];
            b1.y = W1[o + D_FF];
            b2.x = W2[o];
            b2.y = W2[o + D_FF];
            g = __builtin_amdgcn_wmma_f32_16x16x4_f32(false, a, false, b1,
                                                      (short)0, g, false, false);
            v = __builtin_amdgcn_wmma_f32_16x16x4_f32(false, a, false, b2,
                                                      (short)0, v, false, false);
        }

        // ---- silu(gate) * value -> LDS hidden chunk [16][128] ----
#pragma unroll
        for (int vv = 0; vv < 8; ++vv) {
            const float gg = g[vv];
            const float s  = gg / (1.0f + __expf(-gg));
            const float h  = s * v[vv];
            const int   M  = vv + half * 8;               // C/D layout row
            hid[M * H_STRIDE + wave * 16 + lm] = h;
        }
        __syncthreads();

        // ---- phase 2: acc[16 x 1024] += hidden[16 x 128] @ w3[128 x 1024] ----
#pragma unroll
        for (int t = 0; t < 8; ++t) {
            const unsigned n0 = (unsigned)(wave * 128 + t * 16 + lm);
            for (int kk = 0; kk < 128; kk += 4) {
                v2f a;
                a.x = hid[lm * H_STRIDE + kk + kr];
                a.y = hid[lm * H_STRIDE + kk + kr + 1];
                const unsigned o = (unsigned)((fc + kk + kr) * D_MODEL) + n0;
                v2f b3;
                b3.x = W3[o];
                b3.y = W3[o + D_MODEL];
                acc[t] = __builtin_amdgcn_wmma_f32_16x16x4_f32(false, a, false, b3,
                                                               (short)0, acc[t],
                                                               false, false);
            }
        }
        __syncthreads();   // protect hid before next chunk overwrites it
    }

    // ---- store y rows by flat assignment id (unweighted; combine applies w) ----
#pragma unroll
    for (int t = 0; t < 8; ++t) {
        const unsigned n0 = (unsigned)(wave * 128 + t * 16 + lm);
#pragma unroll
        for (int vv = 0; vv < 8; ++vv) {
            const int M  = vv + half * 8;
            const int fa = sfl[M];
            if (fa >= 0) {
                y[(unsigned)(fa * D_MODEL) + n0] = acc[t][vv];
            }
        }
    }
}

// --------------------------- combine ----------------------------------------
__global__ void moe_combine_kernel(const float* __restrict__ y,     // [A, D_MODEL]
                                   const float* __restrict__ ew,    // [N_TOKENS, TOP_K]
                                   float* __restrict__ out)         // [N_TOKENS, D_MODEL]
{
    const int i = blockIdx.x * 256 + threadIdx.x;
    if (i >= N_TOKENS * D_MODEL) return;
    const int t = i >> 10;
    const int d = i & (D_MODEL - 1);
    const float r = ew[t * TOP_K + 0] * y[(unsigned)((t * TOP_K + 0) * D_MODEL + d)]
                  + ew[t * TOP_K + 1] * y[(unsigned)((t * TOP_K + 1) * D_MODEL + d)];
    out[i] = r;
}

// --------------------------- launch ------------------------------------------
extern "C" void kernel_launch(void* const* d_in, const int* in_sizes, int n_in,
                              void* d_out, int out_size, void* d_ws, size_t ws_size,
                              hipStream_t stream) {
    const float* x   = (const float*)d_in[0];
    const int*   ei  = (const int*)  d_in[1];
    const float* ew  = (const float*)d_in[2];
    const float* w1  = (const float*)d_in[3];
    const float* w2  = (const float*)d_in[4];
    const float* w3  = (const float*)d_in[5];
    float*       out = (float*)d_out;

    // Workspace layout
    float* y = (float*)d_ws;                                   // 16 MB
    int*   ip = (int*)(y + (size_t)N_ASSIGN * D_MODEL);
    int* s_tok      = ip; ip += N_ASSIGN;
    int* s_flat     = ip; ip += N_ASSIGN;
    int* tile_e     = ip; ip += MAX_TILES;
    int* tile_srow  = ip; ip += MAX_TILES;
    int* tile_nrows = ip; ip += MAX_TILES;
    int* n_tiles    = ip; ip += 4;

    moe_route_kernel<<<1, 256, 0, stream>>>(ei, s_tok, s_flat,
                                            tile_e, tile_srow, tile_nrows, n_tiles);

    moe_expert_kernel<<<MAX_TILES, 256, LDS_BYTES, stream>>>(
        x, w1, w2, w3, s_tok, s_flat, tile_e, tile_srow, tile_nrows, n_tiles, y);

    moe_combine_kernel<<<(N_TOKENS * D_MODEL + 255) / 256, 256, 0, stream>>>(y, ew, out);

    (void)in_sizes; (void)n_in; (void)out_size; (void)ws_size;
}